// KVCache_83408264888538
// MI455X (gfx1250) — compile-verified
//
#include <hip/hip_runtime.h>
#include <hip/hip_bf16.h>

// KV-cache quantized append + full dequantize for MI455X (gfx1250).
//
// Bandwidth-bound streaming op: ~403 MB of traffic -> ~17.3 us at 23.3 TB/s.
// No matrix math, so no WMMA; the optimized path is b128 non-temporal
// loads/stores (avoid polluting 192MB L2 with a use-once 400MB stream),
// global_prefetch_b8 for the next grid-stride iteration, and a wave32
// shuffle-reduction for the tiny per-row max-abs quantization step.

typedef float f4 __attribute__((ext_vector_type(4)));

#define MAX_INT8 127.5f
#define INV_MAX8 (1.0f / 127.5f)

// ---------------------------------------------------------------------------
// Kernel 1: dequantize the whole stored cache (S,H,B,D) for both K and V.
// One float4 per thread per iteration; 32 threads cover one (s,h,b) row of
// D=128 floats, all sharing one scale. Grid-stride loop with NT hints and
// prefetch of the next iteration's lines.
// ---------------------------------------------------------------------------
__global__ void kv_dequant_bulk(const float* __restrict__ kc,
                                const float* __restrict__ ksc,
                                const float* __restrict__ vc,
                                const float* __restrict__ vsc,
                                float* __restrict__ ok,
                                float* __restrict__ ov,
                                long nquads)   // total float4s per array = S*H*B*32
{
    const long stride = (long)gridDim.x * blockDim.x;
    for (long t = (long)blockIdx.x * blockDim.x + threadIdx.x; t < nquads; t += stride) {
        const long off = t << 2;          // float offset of the t-th float4
        const int  row = (int)(t >> 5);   // (s,h,b) row index: 32 float4s per row

        // gfx1250 global_prefetch_b8 of the next grid-stride iteration
        const long t2 = t + stride;
        if (t2 < nquads) {
            __builtin_prefetch(kc + (t2 << 2), 0, 0);
            __builtin_prefetch(vc + (t2 << 2), 0, 0);
        }

        const float mk = ksc[row] * INV_MAX8;   // scale * (1/127.5)
        const float mv = vsc[row] * INV_MAX8;

        f4 a = __builtin_nontemporal_load((const f4*)(kc + off));
        f4 b = __builtin_nontemporal_load((const f4*)(vc + off));

        __builtin_nontemporal_store(a * mk, (f4*)(ok + off));
        __builtin_nontemporal_store(b * mv, (f4*)(ov + off));
    }
}

// ---------------------------------------------------------------------------
// Kernel 2: quantize the new decode step (logical (B,1,H,D)) and write its
// dequantized value into the output at stored position (idx, h, b, :).
// One wave32 per (array, b, h) row: lane l holds float4 at d = 4*l.
// Runs AFTER kernel 1 on the same stream, so it overwrites the stale s==idx
// rows -> matches dynamic_update_slice semantics deterministically.
// ---------------------------------------------------------------------------
__global__ void kv_quant_step(const float* __restrict__ knew,
                              const float* __restrict__ vnew,
                              float* __restrict__ ok,
                              float* __restrict__ ov,
                              const int* __restrict__ idxp)
{
    const int tid  = blockIdx.x * blockDim.x + threadIdx.x;
    const int row  = tid >> 5;            // 0..127: [0,64)=key rows, [64,128)=value rows
    const int lane = tid & 31;
    if (row >= 128) return;

    const int  bh  = row & 63;            // logical (B,1,H,D): flat index (b*16 + h)
    const int  b   = bh >> 4;
    const int  h   = bh & 15;
    const bool isV = (row >= 64);

    const float* __restrict__ src = isV ? vnew : knew;
    float*       __restrict__ dst = isV ? ov   : ok;

    f4 x = *(const f4*)(src + bh * 128 + lane * 4);

    // per-row max |x| over D=128 via wave32 butterfly reduction
    float m = fmaxf(fmaxf(fabsf(x.x), fabsf(x.y)), fmaxf(fabsf(x.z), fabsf(x.w)));
#pragma unroll
    for (int o = 16; o > 0; o >>= 1)
        m = fmaxf(m, __shfl_xor(m, o, 32));

    // q = rint(x * (127.5/scale)); deq = q * (scale/127.5)  (matches reference order)
    const float r  = MAX_INT8 / m;
    const float dq = m * INV_MAX8;
    f4 q;
    q.x = rintf(x.x * r) * dq;
    q.y = rintf(x.y * r) * dq;
    q.z = rintf(x.z * r) * dq;
    q.w = rintf(x.w * r) * dq;

    // stored layout (S,H,B,D): element offset ((idx*16 + h)*4 + b)*128 + 4*lane
    const int  idx = idxp[0];
    const long o4  = ((long)(idx * 16 + h) * 4 + b) * 128 + lane * 4;
    __builtin_nontemporal_store(q, (f4*)(dst + o4));
}

extern "C" void kernel_launch(void* const* d_in, const int* in_sizes, int n_in,
                              void* d_out, int out_size, void* d_ws, size_t ws_size,
                              hipStream_t stream) {
    const float* knew = (const float*)d_in[0];   // (B,1,H,D)
    const float* vnew = (const float*)d_in[1];   // (B,1,H,D)
    const float* kc   = (const float*)d_in[2];   // (S,H,B,D)
    const float* vc   = (const float*)d_in[3];   // (S,H,B,D)
    const float* ksc  = (const float*)d_in[4];   // (S,H,B,1)
    const float* vsc  = (const float*)d_in[5];   // (S,H,B,1)
    const int*   idx  = (const int*)d_in[6];     // scalar ar_cache_index

    const long n = (long)in_sizes[2];            // S*H*B*D elements per cache
    float* ok = (float*)d_out;                   // k_deq
    float* ov = ok + n;                          // v_deq

    const long nquads = n >> 2;                  // float4s per array

    // Bulk stream: 6144 blocks x 256 threads -> ~4 grid-stride iterations per
    // thread at S=3072, enough in-flight b128 requests to saturate 23.3 TB/s.
    kv_dequant_bulk<<<6144, 256, 0, stream>>>(kc, ksc, vc, vsc, ok, ov, nquads);

    // New-step quantize + scatter overwrite of the s==idx slice (128 wave32s).
    kv_quant_step<<<16, 256, 0, stream>>>(knew, vnew, ok, ov, idx);
}